// IncrementalHyperbolicSegmentationModule_31379031064686
// MI455X (gfx1250) — compile-verified
//
#include <hip/hip_runtime.h>
#include <math.h>

typedef __attribute__((ext_vector_type(2))) float v2f;
typedef __attribute__((ext_vector_type(4))) float v4f;
typedef __attribute__((ext_vector_type(8))) float v8f;

#define POS_TOTAL 32768   // 8 * 64 * 64
#define HW        4096    // 64*64
#define IN_DIM    256
#define EMB       64
#define NCLS      150
#define NCLS_PAD  160
#define NBATCH    8

__device__ __forceinline__ v8f wmma4(v2f a, v2f b, v8f c) {
    // V_WMMA_F32_16X16X4_F32 : D(16x16,f32) = A(16x4,f32) x B(4x16,f32) + C
    return __builtin_amdgcn_wmma_f32_16x16x4_f32(
        false, a, false, b, (short)0, c, false, false);
}

// ---------------------------------------------------------------------------
// Kernel 1: 1x1-conv projection (WMMA f32) + visual_alpha scale + expmap0 +
// ball projection.  Output z stored d-major: zws[d*32768 + pos].
// A = proj_w (M=dims), B = h (K=in_ch x N=positions).
// ---------------------------------------------------------------------------
__global__ __launch_bounds__(256) void k_proj_expmap(
    const float* __restrict__ h, const float* __restrict__ W,
    const float* __restrict__ alpha_p, const float* __restrict__ curv_p,
    float* __restrict__ zws)
{
    __shared__ v2f Wlds[64 * 4 * 32];   // [kk][mt][lane] -> 64 KB

    const int tid = threadIdx.x;
    // Stage proj_w into WMMA A-layout: lane L holds M=(L&15), K = 4*kk + 2*(L>>4) {+0,+1}
    for (int e = tid; e < 64 * 4 * 32; e += 256) {
        int kk = e >> 7, mt = (e >> 5) & 3, ln = e & 31;
        int d = mt * 16 + (ln & 15);
        int i = kk * 4 + ((ln >> 4) << 1);
        Wlds[e] = *(const v2f*)(W + d * IN_DIM + i);
    }
    __syncthreads();

    const float ascale = expf(fminf(alpha_p[0], 0.0f));
    const float kc   = curv_p[0];
    const float sqc  = sqrtf(kc);

    const int wave = tid >> 5, lane = tid & 31;
    const int m = lane & 15, hi = lane >> 4;
    const int pos0 = (blockIdx.x * 8 + wave) << 4;   // 16 positions per wave
    const int bb = pos0 >> 12, hw0 = pos0 & 4095;    // tile never crosses batch

    const float* hb = h + (size_t)bb * IN_DIM * HW + hw0 + m;

    v8f acc0 = {}, acc1 = {}, acc2 = {}, acc3 = {};
    #pragma unroll 8
    for (int kk = 0; kk < 64; ++kk) {
        int i0 = kk * 4 + hi * 2;
        v2f bv;
        bv.x = hb[(size_t)i0 * HW];          // B layout: vgpr0 = K 2*hi
        bv.y = hb[(size_t)(i0 + 1) * HW];    //           vgpr1 = K 2*hi+1
        int base = kk * 128 + lane;
        acc0 = wmma4(Wlds[base      ], bv, acc0);
        acc1 = wmma4(Wlds[base +  32], bv, acc1);
        acc2 = wmma4(Wlds[base +  64], bv, acc2);
        acc3 = wmma4(Wlds[base +  96], bv, acc3);
    }

    // alpha scale + squared norm over all 64 dims of this lane's position
    float part = 0.0f;
    #pragma unroll
    for (int r = 0; r < 8; ++r) {
        acc0[r] *= ascale; acc1[r] *= ascale; acc2[r] *= ascale; acc3[r] *= ascale;
        part += acc0[r] * acc0[r] + acc1[r] * acc1[r]
              + acc2[r] * acc2[r] + acc3[r] * acc3[r];
    }
    part += __shfl_xor(part, 16, 32);   // lanes L, L^16 = same position, other dim-half

    float un = fmaxf(sqrtf(part), 1e-15f);
    float sc = tanhf(sqc * un) / (sqc * un);          // expmap0 scale
    float maxn = (1.0f - 4e-3f) / sqc;                // PROJ_EPS
    float xn = fmaxf(un * sc, 1e-15f);
    if (xn > maxn) sc *= maxn / xn;                   // project onto ball

    float* zb = zws + pos0 + m;
    #pragma unroll
    for (int r = 0; r < 8; ++r) {
        int dr = r + hi * 8;                          // C layout: M = r + 8*hi (+ 16*mt)
        zb[(size_t)(dr     ) * POS_TOTAL] = acc0[r] * sc;
        zb[(size_t)(dr + 16) * POS_TOTAL] = acc1[r] * sc;
        zb[(size_t)(dr + 32) * POS_TOTAL] = acc2[r] * sc;
        zb[(size_t)(dr + 48) * POS_TOTAL] = acc3[r] * sc;
    }
}

// ---------------------------------------------------------------------------
// Kernel 2: Poincare distance to 150 class hyperplanes via two WMMA GEMMs
// (pz = z·p, zw = z·w_hat) + arcsinh formula + logit scale.
// M = classes (10 tiles of 16, padded to 160), N = 16 positions per wave.
// Output cls[(b*150+c)*4096 + hw].
// ---------------------------------------------------------------------------
__global__ __launch_bounds__(256) void k_dist(
    const float* __restrict__ zws, const float* __restrict__ offset,
    const float* __restrict__ normal, const float* __restrict__ curv_p,
    const float* __restrict__ lsc_p, float* __restrict__ cls)
{
    __shared__ v2f  Ap[16 * 10 * 32];    // p = -offset, WMMA A-layout, 40 KB
    __shared__ v2f  Aw[16 * 10 * 32];    // w_hat,       WMMA A-layout, 40 KB
    __shared__ float p2s[NCLS_PAD], pws[NCLS_PAD], w2s[NCLS_PAD];

    const int tid = threadIdx.x;
    const float kc = curv_p[0];

    if (tid < NCLS_PAD) {
        const int c = tid;
        float off2 = 0.0f, n2 = 0.0f, don = 0.0f;
        if (c < NCLS) {
            const float* oc = offset + c * EMB;
            const float* nc = normal + c * EMB;
            #pragma unroll 8
            for (int d = 0; d < EMB; ++d) {
                float o = oc[d], n = nc[d];
                off2 += o * o; n2 += n * n; don += o * n;
            }
        }
        float conf = 1.0f - kc * off2;                  // conformal factor
        float w2   = fabsf(conf) * sqrtf(n2);           // ||bias||
        float invw = conf / fmaxf(w2, 1e-12f);          // w_hat = normal * invw
        float pw   = -don * invw;                       // (-offset)·w_hat
        if (c >= NCLS) { off2 = 0.0f; pw = 0.0f; w2 = 0.0f; invw = 0.0f; }
        p2s[c] = off2; pws[c] = pw; w2s[c] = w2;

        const int ct = c >> 4, mm = c & 15;
        for (int kk = 0; kk < 16; ++kk) {
            #pragma unroll
            for (int h2 = 0; h2 < 2; ++h2) {
                int lanei = mm + 16 * h2;
                int k = kk * 4 + 2 * h2;
                int idx = (kk * 10 + ct) * 32 + lanei;
                v2f ov = {}, nv = {};
                if (c < NCLS) {
                    ov = *(const v2f*)(offset + c * EMB + k);
                    nv = *(const v2f*)(normal + c * EMB + k);
                }
                v2f pv; pv.x = -ov.x;       pv.y = -ov.y;
                v2f wv; wv.x = nv.x * invw; wv.y = nv.y * invw;
                Ap[idx] = pv; Aw[idx] = wv;
            }
        }
    }
    __syncthreads();

    const int wave = tid >> 5, lane = tid & 31;
    const int m = lane & 15, hi = lane >> 4;
    const int pos0 = (blockIdx.x * 8 + wave) << 4;
    const int bb = pos0 >> 12, hw0 = pos0 & 4095;

    // z B-operands: load once, reuse across all 10 class tiles
    v2f zb[16];
    float part = 0.0f;
    const float* zp = zws + pos0 + m;
    #pragma unroll
    for (int kk = 0; kk < 16; ++kk) {
        int k = kk * 4 + 2 * hi;
        v2f v;
        v.x = zp[(size_t)k * POS_TOTAL];
        v.y = zp[(size_t)(k + 1) * POS_TOTAL];
        zb[kk] = v;
        part += v.x * v.x + v.y * v.y;
    }
    const float z2 = part + __shfl_xor(part, 16, 32);   // ||z||^2, per-lane position

    const float sqk  = sqrtf(kc);
    const float ls   = expf(fminf(lsc_p[0], 4.6052f));
    const float maxn = (1.0f - 0.001f) / sqk;           // TEMP
    const float mn2  = maxn * maxn;

    for (int ct = 0; ct < 10; ++ct) {
        v8f pz = {}, zw = {};
        #pragma unroll
        for (int kk = 0; kk < 16; ++kk) {
            int base = (kk * 10 + ct) * 32 + lane;
            pz = wmma4(Ap[base], zb[kk], pz);
            zw = wmma4(Aw[base], zb[kk], zw);
        }
        #pragma unroll
        for (int r = 0; r < 8; ++r) {
            int c = ct * 16 + r + 8 * hi;               // C layout: class row
            float p2 = p2s[c], pw = pws[c], w2 = w2s[c];
            float pzv = pz[r], zwv = zw[r];
            float d1 = 1.0f + 2.0f * kc * pzv + kc * kc * z2 * p2;
            float rd = 1.0f / d1;
            float av = (1.0f + 2.0f * kc * pzv + kc * z2) * rd;
            float bv = (1.0f - kc * p2) * rd;
            float sq = av * av * p2 + 2.0f * av * bv * pzv + bv * bv * z2;
            float sqn = sqrtf(sq);
            float np  = (sqn < maxn) ? sq : mn2;
            float psc = (sqn > maxn) ? maxn / fmaxf(sqn, 1e-15f) : 1.0f;
            float nom = 2.0f * sqk * ((av * pw + bv * zwv) * psc);
            float den = fmaxf(1.0f - kc * np, 1e-15f);
            float dist = 2.0f * w2 / sqk * asinhf(nom / den);
            if (c < NCLS)
                cls[((size_t)(bb * NCLS + c)) * HW + hw0 + m] = dist * ls;
        }
    }
}

// ---------------------------------------------------------------------------
// Kernel 3: 4x bilinear upsample 64x64 -> 256x256 (half-pixel, edge clamp).
// Fixed-scale specialization: each thread computes an aligned quad of 4
// output pixels (input cols {ix0-1, ix0, ix0+1}, weights {.625,.875,.125,.375})
// -> 6 loads + one b128 store.  4x fewer waves/stores than the b32 version.
// ---------------------------------------------------------------------------
__global__ __launch_bounds__(256) void k_resize(
    const float* __restrict__ cls, float* __restrict__ out)
{
    const int p   = blockIdx.y;                    // plane: b*150 + c  (1200)
    const int tid = threadIdx.x;
    const int oy  = blockIdx.x * 4 + (tid >> 6);   // 4 output rows per block
    const int ix0 = tid & 63;                      // quad -> input column
    const int ox0 = ix0 << 2;                      // output x of quad start

    // y interpolation (generic half-pixel)
    float fy = oy * 0.25f - 0.375f;
    float y0f = floorf(fy);
    float wy = fy - y0f;
    int y0 = (int)y0f;
    int y0c = y0 < 0 ? 0 : (y0 > 63 ? 63 : y0);
    int y1c = y0 + 1 > 63 ? 63 : (y0 + 1 < 0 ? 0 : y0 + 1);

    // x taps: cm = ix0-1 (clamped), c0 = ix0, cp = ix0+1 (clamped)
    int cm = ix0 > 0 ? ix0 - 1 : 0;
    int cp = ix0 < 63 ? ix0 + 1 : 63;

    const float* r0 = cls + (size_t)p * HW + y0c * 64;
    const float* r1 = cls + (size_t)p * HW + y1c * 64;
    float am = r0[cm], a0 = r0[ix0], ap = r0[cp];
    float bm = r1[cm], b0 = r1[ix0], bp = r1[cp];

    // blend rows first (common wy), then the 4 fixed x-weights
    float vm = am + wy * (bm - am);
    float v0 = a0 + wy * (b0 - a0);
    float vp = ap + wy * (bp - ap);

    v4f q;
    q.x = vm + 0.625f * (v0 - vm);
    q.y = vm + 0.875f * (v0 - vm);
    q.z = v0 + 0.125f * (vp - v0);
    q.w = v0 + 0.375f * (vp - v0);

    *(v4f*)(out + (size_t)p * 65536 + oy * 256 + ox0) = q;
}

// ---------------------------------------------------------------------------
extern "C" void kernel_launch(void* const* d_in, const int* in_sizes, int n_in,
                              void* d_out, int out_size, void* d_ws, size_t ws_size,
                              hipStream_t stream) {
    (void)in_sizes; (void)n_in; (void)out_size; (void)ws_size;
    const float* h      = (const float*)d_in[0];  // (8,256,64,64)
    const float* W      = (const float*)d_in[1];  // (64,256)
    const float* offset = (const float*)d_in[2];  // (150,64)
    const float* normal = (const float*)d_in[3];  // (150,64)
    const float* alpha  = (const float*)d_in[4];  // scalar
    const float* lsc    = (const float*)d_in[5];  // scalar
    const float* curv   = (const float*)d_in[6];  // scalar
    // d_in[7]/d_in[8] = out_h/out_w = 256 (hardcoded)

    float* zws = (float*)d_ws;                          // (64, 32768) d-major, 8 MB
    float* cl  = zws + (size_t)EMB * POS_TOTAL;         // (1200, 4096), ~19.7 MB
    float* out = (float*)d_out;                         // (8,150,256,256)

    k_proj_expmap<<<dim3(POS_TOTAL / 16 / 8), 256, 0, stream>>>(h, W, alpha, curv, zws);
    k_dist       <<<dim3(POS_TOTAL / 16 / 8), 256, 0, stream>>>(zws, offset, normal, curv, lsc, cl);
    k_resize     <<<dim3(64, NBATCH * NCLS), 256, 0, stream>>>(cl, out);
}